// MultiHeadAttention_53094385713786
// MI455X (gfx1250) — compile-verified
//
#include <hip/hip_runtime.h>
#include <math.h>

// ---------------------------------------------------------------------------
// Graph-transformer layer for MI455X (gfx1250, wave32, WMMA).
// GEMMs use V_WMMA_F32_16X16X4_F32 (full fp32 precision; compute is trivially
// small vs the 23.3 TB/s memory system, so no need to downcast).
// Edge softmax/scatter phase dominates; it is kept L2-resident (192 MB L2).
// ---------------------------------------------------------------------------

typedef float v2f __attribute__((ext_vector_type(2)));
typedef float v8f __attribute__((ext_vector_type(8)));

constexpr int D  = 128;   // model dim
constexpr int H  = 8;     // heads
constexpr int DH = 16;    // head dim
constexpr int DF = 512;   // FFN dim

#if defined(__gfx1250__) && __has_builtin(__builtin_amdgcn_wmma_f32_16x16x4_f32)
#define USE_WMMA_F32X4 1
#else
#define USE_WMMA_F32X4 0
#endif

__device__ __forceinline__ void atomic_max_float(float* addr, float val) {
    // standard monotonic bit-pattern trick; works with -inf init
    if (val >= 0.0f) atomicMax((int*)addr, __float_as_int(val));
    else             atomicMin((unsigned int*)addr, __float_as_uint(val));
}

// ---------------------------------------------------------------------------
// Generic fp32 WMMA GEMM:  out[m][n] = sum_k X[m][k] * W[n][k]  (+bias)(+res)(relu)
// Block = 128 threads (4 waves). M-tile = 16 rows (staged in LDS, padded to
// kill bank conflicts). Each wave computes two 16x16 N-tiles => block covers
// a 16 x 128 output strip. grid = (M/16, N/128).
// ---------------------------------------------------------------------------
__global__ __launch_bounds__(128)
void gemm_wmma_f32(const float* __restrict__ X, const float* __restrict__ W,
                   const float* __restrict__ bias, const float* __restrict__ res,
                   float* __restrict__ out, int K, int N, int relu)
{
    __shared__ float As[16 * (512 + 4)];     // supports K up to 512
    const int ldk  = K + 4;                  // pad rows: stride % 64 banks != 0
    const int m0   = blockIdx.x * 16;
    const int tid  = threadIdx.x;
    const int wave = tid >> 5;
    const int lane = tid & 31;
    const int l    = lane & 15;              // N / M sub-index
    const int hi   = lane >> 4;              // half-wave selector

    // cooperative float4 load of the 16 x K A-tile
    const int kv4 = K >> 2;
    for (int i = tid; i < 16 * kv4; i += 128) {
        int r = i / kv4;
        int c = (i - r * kv4) << 2;
        float4 t = *(const float4*)&X[(size_t)(m0 + r) * K + c];
        *(float4*)&As[r * ldk + c] = t;
    }
    __syncthreads();

    const int n0 = blockIdx.y * 128 + wave * 32;
    const float* Wp0 = W + (size_t)(n0 + l) * K;        // B col n0+l
    const float* Wp1 = W + (size_t)(n0 + 16 + l) * K;   // B col n0+16+l
    const float* Ap  = As + l * ldk;

    v8f c0 = {};
    v8f c1 = {};
    for (int k0 = 0; k0 < K; k0 += 4) {
        const int ko = k0 + 2 * hi;          // half-wave picks K pair (ISA 16x4 layout)
#if USE_WMMA_F32X4
        float2 af  = *(const float2*)&Ap[ko];
        float2 b0f = *(const float2*)&Wp0[ko];
        float2 b1f = *(const float2*)&Wp1[ko];
        v2f a;  a.x  = af.x;  a.y  = af.y;
        v2f b0; b0.x = b0f.x; b0.y = b0f.y;
        v2f b1; b1.x = b1f.x; b1.y = b1f.y;
        c0 = __builtin_amdgcn_wmma_f32_16x16x4_f32(false, a, false, b0, (short)0, c0, false, false);
        c1 = __builtin_amdgcn_wmma_f32_16x16x4_f32(false, a, false, b1, (short)0, c1, false, false);
#else
        (void)ko;
        #pragma unroll
        for (int kk = 0; kk < 4; ++kk) {
            float bb0 = Wp0[k0 + kk];
            float bb1 = Wp1[k0 + kk];
            #pragma unroll
            for (int r = 0; r < 8; ++r) {
                float av = As[(r + 8 * hi) * ldk + k0 + kk];
                c0[r] += av * bb0;
                c1[r] += av * bb1;
            }
        }
#endif
    }

    // D layout: VGPR r -> row m0 + r + 8*hi ; lane l -> col n0 + l (tile0), +16 (tile1)
    #pragma unroll
    for (int r = 0; r < 8; ++r) {
        const int m  = m0 + r + 8 * hi;
        const int n  = n0 + l;
        float x0 = c0[r];
        float x1 = c1[r];
        if (bias) { x0 += bias[n]; x1 += bias[n + 16]; }
        if (res)  { x0 += res[(size_t)m * N + n]; x1 += res[(size_t)m * N + n + 16]; }
        if (relu) { x0 = fmaxf(x0, 0.0f); x1 = fmaxf(x1, 0.0f); }
        out[(size_t)m * N + n]      = x0;
        out[(size_t)m * N + n + 16] = x1;
    }
}

// ---------------------------------------------------------------------------
// Pass 0: init max=-inf, denom=0, attout=0
// ---------------------------------------------------------------------------
__global__ void init_kernel(float* __restrict__ mx, float* __restrict__ dn,
                            float* __restrict__ ao, int nmh, int nao)
{
    int i = blockIdx.x * blockDim.x + threadIdx.x;
    if (i < nmh) { mx[i] = -INFINITY; dn[i] = 0.0f; }
    if (i < nao) ao[i] = 0.0f;
}

// ---------------------------------------------------------------------------
// Pass 1: per (edge, head) score = (k[src] . q[dst]) / sqrt(DH); atomic max
// ---------------------------------------------------------------------------
__global__ void edge_score_kernel(const float* __restrict__ q, const float* __restrict__ k,
                                  const int* __restrict__ src, const int* __restrict__ dst,
                                  float* __restrict__ ex, float* __restrict__ mx, int E)
{
    int i = blockIdx.x * blockDim.x + threadIdx.x;
    if (i >= E * H) return;
    int e = i >> 3, h = i & 7;
    int s = src[e], d = dst[e];
    const float4* kp = (const float4*)(k + (size_t)s * D + h * DH);
    const float4* qp = (const float4*)(q + (size_t)d * D + h * DH);
    float acc = 0.0f;
    #pragma unroll
    for (int j = 0; j < 4; ++j) {
        float4 a = kp[j], b = qp[j];
        acc += a.x * b.x + a.y * b.y + a.z * b.z + a.w * b.w;
    }
    float sc = acc * 0.25f;  // 1/sqrt(16)
    ex[i] = sc;
    atomic_max_float(&mx[d * H + h], sc);
}

// ---------------------------------------------------------------------------
// Pass 2: ex = exp(score - max[dst]); atomic sum into denom
// ---------------------------------------------------------------------------
__global__ void edge_exp_kernel(const int* __restrict__ dst, float* __restrict__ ex,
                                const float* __restrict__ mx, float* __restrict__ dn, int E)
{
    int i = blockIdx.x * blockDim.x + threadIdx.x;
    if (i >= E * H) return;
    int e = i >> 3, h = i & 7;
    int d = dst[e];
    float val = expf(ex[i] - mx[d * H + h]);
    ex[i] = val;
    atomicAdd(&dn[d * H + h], val);
}

// ---------------------------------------------------------------------------
// Pass 3: attout[dst] += v[src] * (ex / denom[dst]); 32 lanes per edge
// lane -> head = lane/4, 4 contiguous dims per lane (float4 gather of v)
// ---------------------------------------------------------------------------
__global__ void edge_scatter_kernel(const float* __restrict__ v,
                                    const int* __restrict__ src, const int* __restrict__ dst,
                                    const float* __restrict__ ex, const float* __restrict__ dn,
                                    float* __restrict__ ao, int E)
{
    int j = blockIdx.x * blockDim.x + threadIdx.x;
    if (j >= E * 32) return;
    int e  = j >> 5, ln = j & 31;
    int h  = ln >> 2;
    int d0 = (ln & 3) * 4;
    int s = src[e], d = dst[e];
    float att = ex[e * H + h] / dn[d * H + h];
    float4 vv = *(const float4*)(v + (size_t)s * D + h * DH + d0);
    float* op = ao + (size_t)d * D + h * DH + d0;
    atomicAdd(op + 0, vv.x * att);
    atomicAdd(op + 1, vv.y * att);
    atomicAdd(op + 2, vv.z * att);
    atomicAdd(op + 3, vv.w * att);
}

// ---------------------------------------------------------------------------
// LayerNorm over D=128: one wave32 per row, 8 rows per 256-thread block
// ---------------------------------------------------------------------------
__global__ __launch_bounds__(256)
void layernorm_kernel(const float* __restrict__ x, const float* __restrict__ g,
                      const float* __restrict__ b, float* __restrict__ out, int M)
{
    int row  = blockIdx.x * 8 + (threadIdx.x >> 5);
    int lane = threadIdx.x & 31;
    if (row >= M) return;
    float4 xv = *(const float4*)(x + (size_t)row * D + lane * 4);
    float s  = xv.x + xv.y + xv.z + xv.w;
    float ss = xv.x * xv.x + xv.y * xv.y + xv.z * xv.z + xv.w * xv.w;
    #pragma unroll
    for (int o = 16; o > 0; o >>= 1) {
        s  += __shfl_xor(s,  o, 32);
        ss += __shfl_xor(ss, o, 32);
    }
    float mu  = s * (1.0f / 128.0f);
    float var = ss * (1.0f / 128.0f) - mu * mu;
    float rs  = rsqrtf(var + 1e-5f);
    float4 gv = *(const float4*)(g + lane * 4);
    float4 bv = *(const float4*)(b + lane * 4);
    float4 ov;
    ov.x = (xv.x - mu) * rs * gv.x + bv.x;
    ov.y = (xv.y - mu) * rs * gv.y + bv.y;
    ov.z = (xv.z - mu) * rs * gv.z + bv.z;
    ov.w = (xv.w - mu) * rs * gv.w + bv.w;
    *(float4*)(out + (size_t)row * D + lane * 4) = ov;
}

// ---------------------------------------------------------------------------
extern "C" void kernel_launch(void* const* d_in, const int* in_sizes, int n_in,
                              void* d_out, int out_size, void* d_ws, size_t ws_size,
                              hipStream_t stream)
{
    const float* q_feat  = (const float*)d_in[0];
    const float* kv_feat = (const float*)d_in[1];
    const int*   src     = (const int*)  d_in[2];
    const int*   dst     = (const int*)  d_in[3];
    const float* Wq      = (const float*)d_in[4];
    const float* Wk      = (const float*)d_in[5];
    const float* Wv      = (const float*)d_in[6];
    const float* Wo      = (const float*)d_in[7];
    const float* W1      = (const float*)d_in[8];
    const float* bf1     = (const float*)d_in[9];
    const float* W2      = (const float*)d_in[10];
    const float* bf2     = (const float*)d_in[11];
    const float* ln1_g   = (const float*)d_in[12];
    const float* ln1_b   = (const float*)d_in[13];
    const float* ln2_g   = (const float*)d_in[14];
    const float* ln2_b   = (const float*)d_in[15];
    float* out = (float*)d_out;

    const int NQ  = in_sizes[0] / D;   // 20000
    const int NKV = in_sizes[1] / D;   // 20000
    const int E   = in_sizes[2];       // 640000

    // workspace layout (floats); hidden aliases the dead q/k/v/ex region
    float* ws = (float*)d_ws;
    size_t o_q    = 0;
    size_t o_k    = o_q  + (size_t)NQ  * D;
    size_t o_v    = o_k  + (size_t)NKV * D;
    size_t o_ex   = o_v  + (size_t)NKV * D;
    size_t regA   = o_ex + (size_t)E * H;            // 12.8M floats
    size_t o_mx   = regA;
    size_t o_dn   = o_mx + (size_t)NQ * H;
    size_t o_ao   = o_dn + (size_t)NQ * H;
    size_t o_tmp  = o_ao + (size_t)NQ * D;
    size_t o_feat = o_tmp + (size_t)NQ * D;

    float* q      = ws + o_q;
    float* k      = ws + o_k;
    float* v      = ws + o_v;
    float* ex     = ws + o_ex;
    float* hidden = ws + 0;            // alias: q/k/v/ex dead before FFN1
    float* mx     = ws + o_mx;
    float* dn     = ws + o_dn;
    float* ao     = ws + o_ao;
    float* tmp    = ws + o_tmp;
    float* feat   = ws + o_feat;

    const dim3 blk(128);

    // 0) init softmax state + attention accumulator
    {
        int nmh = NQ * H, nao = NQ * D;
        int n = (nao > nmh) ? nao : nmh;
        init_kernel<<<(n + 255) / 256, 256, 0, stream>>>(mx, dn, ao, nmh, nao);
    }

    // 1) Q/K/V projections (WMMA fp32)
    gemm_wmma_f32<<<dim3(NQ  / 16, 1), blk, 0, stream>>>(q_feat,  Wq, nullptr, nullptr, q, D, D, 0);
    gemm_wmma_f32<<<dim3(NKV / 16, 1), blk, 0, stream>>>(kv_feat, Wk, nullptr, nullptr, k, D, D, 0);
    gemm_wmma_f32<<<dim3(NKV / 16, 1), blk, 0, stream>>>(kv_feat, Wv, nullptr, nullptr, v, D, D, 0);

    // 2) edge softmax (3 passes, L2-resident)
    {
        int t = E * H;
        edge_score_kernel<<<(t + 255) / 256, 256, 0, stream>>>(q, k, src, dst, ex, mx, E);
        edge_exp_kernel  <<<(t + 255) / 256, 256, 0, stream>>>(dst, ex, mx, dn, E);
        int t2 = E * 32;
        edge_scatter_kernel<<<(t2 + 255) / 256, 256, 0, stream>>>(v, src, dst, ex, dn, ao, E);
    }

    // 3) output projection + residual, then LN1
    gemm_wmma_f32<<<dim3(NQ / 16, 1), blk, 0, stream>>>(ao, Wo, nullptr, q_feat, tmp, D, D, 0);
    layernorm_kernel<<<(NQ + 7) / 8, 256, 0, stream>>>(tmp, ln1_g, ln1_b, feat, NQ);

    // 4) FFN: relu(feat@W1.T + b1) @ W2.T + b2 + feat, then LN2 -> out
    gemm_wmma_f32<<<dim3(NQ / 16, DF / 128), blk, 0, stream>>>(feat, W1, bf1, nullptr, hidden, D, DF, 1);
    gemm_wmma_f32<<<dim3(NQ / 16, 1), blk, 0, stream>>>(hidden, W2, bf2, feat, tmp, DF, D, 0);
    layernorm_kernel<<<(NQ + 7) / 8, 256, 0, stream>>>(tmp, ln2_g, ln2_b, out, NQ);
}